// MessagePassingLayer_54082228191971
// MI455X (gfx1250) — compile-verified
//
#include <hip/hip_runtime.h>

#define N_ATOMS 20000
#define N_EDGES 320000
#define NCH     16
#define NRBF    8
#define NSLOT   13            // 1 (l=0) + 3 (l=1) + 9 (l=2) spatial components
#define CUTOFF_F 4.0f

typedef float v2f __attribute__((ext_vector_type(2)));
typedef float v8f __attribute__((ext_vector_type(8)));

// D(16x16 f32) = A(16x4 f32) x B(4x16 f32) + C   -- native fp32 WMMA on gfx1250
__device__ __forceinline__ v8f wmma4(v2f a, v2f b, v8f c) {
  return __builtin_amdgcn_wmma_f32_16x16x4_f32(
      /*neg_a=*/false, a, /*neg_b=*/false, b,
      /*c_mod=*/(short)0, c, /*reuse_a=*/false, /*reuse_b=*/false);
}

__global__ __launch_bounds__(256) void zero_ws_kernel(float* __restrict__ p, int n) {
  int i = blockIdx.x * blockDim.x + threadIdx.x;
  if (i < n) p[i] = 0.0f;
}

// ---------------------------------------------------------------------------
// Edge kernel: one wave handles a tile of 16 edges.
//   lin1 (16x16 channel linear per spatial component) -> 52 x v_wmma_f32_16x16x4_f32
//   then lane-local radial basis + tensor products, atomic scatter to node_msg.
// node_msg layout: [atom][ch][slot], slot: 0 = l0, 1..3 = l1, 4..12 = l2
// ---------------------------------------------------------------------------
__global__ __launch_bounds__(256) void edge_kernel(
    const int*   __restrict__ eidx,      // [2][N_EDGES]
    const float* __restrict__ pos,       // [N_ATOMS][3]
    const float* __restrict__ feats0,    // [N_ATOMS][16]
    const float* __restrict__ feats1,    // [N_ATOMS][16][3]
    const float* __restrict__ feats2,    // [N_ATOMS][16][9]
    const float* __restrict__ bessel_w,  // [8]
    const float* __restrict__ rbf_w,     // [3][16][8]
    const float* __restrict__ rbf_b,     // [3][16]
    const float* __restrict__ lin1_w,    // [3][16][16]
    const float* __restrict__ lin1_b,    // [16]
    float*       __restrict__ node_msg)  // [N_ATOMS][16][13]
{
  const int lane = threadIdx.x & 31;
  const int wid  = blockIdx.x * (blockDim.x >> 5) + (threadIdx.x >> 5);
  const int e0   = wid * 16;
  if (e0 >= N_EDGES) return;          // wave-uniform

  const int n    = lane & 15;         // edge within tile == WMMA row/col index
  const int half = lane >> 4;         // K half within WMMA operand
  const int e    = e0 + n;
  const int ai   = eidx[e];           // idx_i (scatter target)
  const int aj   = eidx[N_EDGES + e]; // idx_j (gather source)

  // ---- x[l] = lin1_w[l] @ feats[l][aj]  via f32 WMMA ---------------------
  v8f acc[NSLOT];
#pragma unroll
  for (int s = 0; s < NSLOT; ++s) acc[s] = (v8f){0,0,0,0,0,0,0,0};

#pragma unroll
  for (int kk = 0; kk < 4; ++kk) {
    const int c0 = kk * 4 + half * 2;         // this lane's K channels
    v2f aw0, aw1, aw2;                        // A = weight rows (row = n)
    aw0.x = lin1_w[      n*16 + c0]; aw0.y = lin1_w[      n*16 + c0 + 1];
    aw1.x = lin1_w[256 + n*16 + c0]; aw1.y = lin1_w[256 + n*16 + c0 + 1];
    aw2.x = lin1_w[512 + n*16 + c0]; aw2.y = lin1_w[512 + n*16 + c0 + 1];

    {
      v2f b; b.x = feats0[aj*16 + c0]; b.y = feats0[aj*16 + c0 + 1];
      acc[0] = wmma4(aw0, b, acc[0]);
    }
#pragma unroll
    for (int cmp = 0; cmp < 3; ++cmp) {
      v2f b;
      b.x = feats1[(aj*16 + c0    )*3 + cmp];
      b.y = feats1[(aj*16 + c0 + 1)*3 + cmp];
      acc[1 + cmp] = wmma4(aw1, b, acc[1 + cmp]);
    }
#pragma unroll
    for (int cmp = 0; cmp < 9; ++cmp) {
      v2f b;
      b.x = feats2[(aj*16 + c0    )*9 + cmp];
      b.y = feats2[(aj*16 + c0 + 1)*9 + cmp];
      acc[4 + cmp] = wmma4(aw2, b, acc[4 + cmp]);
    }
  }
  // bias only on l=0 (D element v in this lane = out-channel v + 8*half)
#pragma unroll
  for (int v = 0; v < 8; ++v) acc[0][v] += lin1_b[v + 8*half];

  // ---- per-edge geometry + radial basis (lane-local) ---------------------
  const float pix = pos[ai*3+0], piy = pos[ai*3+1], piz = pos[ai*3+2];
  const float rx = pos[aj*3+0]-pix, ry = pos[aj*3+1]-piy, rz = pos[aj*3+2]-piz;
  const float d2   = rx*rx + ry*ry + rz*rz;
  const float dinv = rsqrtf(d2);
  const float d    = d2 * dinv;
  const float u[3] = { rx*dinv, ry*dinv, rz*dinv };

  const float t  = d * (1.0f / CUTOFF_F);
  const float t2 = t*t, t4 = t2*t2, t6 = t4*t2, t7 = t6*t, t8 = t7*t;
  const float poly = 1.0f - 28.0f*t6 + 48.0f*t7 - 21.0f*t8;
  const float env  = (t < 1.0f) ? poly : 0.0f;
  const float pref = 0.70710678118f * env * dinv;   // sqrt(2/CUTOFF)=sqrt(0.5)
  float rbf[NRBF];
#pragma unroll
  for (int r = 0; r < NRBF; ++r) rbf[r] = pref * __sinf(bessel_w[r] * d);

  // fn[l][ch] = rbf . rbf_w[l][ch] + rbf_b[l][ch], for this lane's 8 channels
  float fn0[8], fn1[8], fn2[8];
#pragma unroll
  for (int v = 0; v < 8; ++v) {
    const int ch = v + 8*half;
    float s0 = rbf_b[ch], s1 = rbf_b[16 + ch], s2 = rbf_b[32 + ch];
#pragma unroll
    for (int r = 0; r < NRBF; ++r) {
      const float rv = rbf[r];
      s0 += rv * rbf_w[(     ch)*8 + r];
      s1 += rv * rbf_w[(16 + ch)*8 + r];
      s2 += rv * rbf_w[(32 + ch)*8 + r];
    }
    fn0[v] = s0; fn1[v] = s1; fn2[v] = s2;
  }

  // ---- tensor products (all 11 TP combos) + atomic segment-sum -----------
#pragma unroll
  for (int v = 0; v < 8; ++v) {
    const int ch = v + 8*half;
    const float x0 = acc[0][v];
    float x1[3], x2[9];
#pragma unroll
    for (int a = 0; a < 3; ++a) x1[a] = acc[1 + a][v];
#pragma unroll
    for (int a = 0; a < 9; ++a) x2[a] = acc[4 + a][v];

    const float y0 = fn0[v];
    float y1[3], y2[9];
#pragma unroll
    for (int a = 0; a < 3; ++a) y1[a] = u[a] * fn1[v];
#pragma unroll
    for (int a = 0; a < 3; ++a)
#pragma unroll
      for (int b = 0; b < 3; ++b) y2[a*3+b] = u[a] * u[b] * fn2[v];

    // zw = 0 : (0,0,0) + (1,1,0) + (2,2,0)
    float m0 = x0 * y0;
#pragma unroll
    for (int a = 0; a < 3; ++a) m0 += x1[a] * y1[a];
#pragma unroll
    for (int a = 0; a < 9; ++a) m0 += x2[a] * y2[a];

    // zw = 1 : (0,1,1) + (1,0,1) + (1,2,1) + (2,1,1)
    float m1[3];
#pragma unroll
    for (int a = 0; a < 3; ++a) {
      float s = x0 * y1[a] + x1[a] * y0;
#pragma unroll
      for (int k = 0; k < 3; ++k) s += x1[k]*y2[k*3+a] + x2[a*3+k]*y1[k];
      m1[a] = s;
    }

    // zw = 2 : (0,2,2) + (1,1,2) + (2,0,2) + (2,2,2)
    float m2[9];
#pragma unroll
    for (int a = 0; a < 3; ++a)
#pragma unroll
      for (int b = 0; b < 3; ++b) {
        float s = x0*y2[a*3+b] + x1[a]*y1[b] + x2[a*3+b]*y0;
#pragma unroll
        for (int k = 0; k < 3; ++k) s += x2[a*3+k]*y2[k*3+b];
        m2[a*3+b] = s;
      }

    float* p = node_msg + (ai*16 + ch) * NSLOT;
    atomicAdd(p + 0, m0);                                   // global_atomic_add_f32
#pragma unroll
    for (int a = 0; a < 3; ++a) atomicAdd(p + 1 + a, m1[a]);
#pragma unroll
    for (int a = 0; a < 9; ++a) atomicAdd(p + 4 + a, m2[a]);
  }
}

// ---------------------------------------------------------------------------
// Node kernel: one wave handles a tile of 16 atoms.
//   lin2 via 52 x v_wmma_f32_16x16x4_f32, then gated activations + residual.
// ---------------------------------------------------------------------------
__global__ __launch_bounds__(256) void node_kernel(
    const float* __restrict__ feats0,
    const float* __restrict__ feats1,
    const float* __restrict__ feats2,
    const float* __restrict__ lin2_w,   // [3][16][16]
    const float* __restrict__ lin2_b,   // [16]
    const float* __restrict__ act_w,    // [2][16]
    const float* __restrict__ act_b,    // [2][16]
    const float* __restrict__ node_msg, // [N_ATOMS][16][13]
    float* __restrict__ out0,
    float* __restrict__ out1,
    float* __restrict__ out2)
{
  const int lane = threadIdx.x & 31;
  const int wid  = blockIdx.x * (blockDim.x >> 5) + (threadIdx.x >> 5);
  const int a0   = wid * 16;
  if (a0 >= N_ATOMS) return;          // wave-uniform, before any WMMA

  const int n    = lane & 15;
  const int half = lane >> 4;
  const int atom = a0 + n;

  v8f acc[NSLOT];
#pragma unroll
  for (int s = 0; s < NSLOT; ++s) acc[s] = (v8f){0,0,0,0,0,0,0,0};

#pragma unroll
  for (int kk = 0; kk < 4; ++kk) {
    const int c0 = kk * 4 + half * 2;
    v2f aw0, aw1, aw2;
    aw0.x = lin2_w[      n*16 + c0]; aw0.y = lin2_w[      n*16 + c0 + 1];
    aw1.x = lin2_w[256 + n*16 + c0]; aw1.y = lin2_w[256 + n*16 + c0 + 1];
    aw2.x = lin2_w[512 + n*16 + c0]; aw2.y = lin2_w[512 + n*16 + c0 + 1];
    const float* mpa = node_msg + (atom*16 + c0    ) * NSLOT;
    const float* mpb = node_msg + (atom*16 + c0 + 1) * NSLOT;
#pragma unroll
    for (int s = 0; s < NSLOT; ++s) {
      v2f b; b.x = mpa[s]; b.y = mpb[s];
      const v2f aw = (s == 0) ? aw0 : (s < 4) ? aw1 : aw2;
      acc[s] = wmma4(aw, b, acc[s]);
    }
  }

#pragma unroll
  for (int v = 0; v < 8; ++v) {
    const int ch = v + 8*half;
    // l = 0 : silu + residual
    const float h0  = acc[0][v] + lin2_b[ch];
    const float sg0 = 1.0f / (1.0f + __expf(-h0));
    out0[atom*16 + ch] = feats0[atom*16 + ch] + h0 * sg0;

    // l = 1 : sigmoid(norm)-gated + residual
    float h1[3], ss1 = 0.0f;
#pragma unroll
    for (int a = 0; a < 3; ++a) { h1[a] = acc[1 + a][v]; ss1 += h1[a]*h1[a]; }
    const float g1 = 1.0f / (1.0f + __expf(-(act_w[ch] * ss1 + act_b[ch])));
#pragma unroll
    for (int a = 0; a < 3; ++a)
      out1[(atom*16 + ch)*3 + a] = feats1[(atom*16 + ch)*3 + a] + g1 * h1[a];

    // l = 2
    float h2[9], ss2 = 0.0f;
#pragma unroll
    for (int a = 0; a < 9; ++a) { h2[a] = acc[4 + a][v]; ss2 += h2[a]*h2[a]; }
    const float g2 = 1.0f / (1.0f + __expf(-(act_w[16 + ch] * ss2 + act_b[16 + ch])));
#pragma unroll
    for (int a = 0; a < 9; ++a)
      out2[(atom*16 + ch)*9 + a] = feats2[(atom*16 + ch)*9 + a] + g2 * h2[a];
  }
}

extern "C" void kernel_launch(void* const* d_in, const int* in_sizes, int n_in,
                              void* d_out, int out_size, void* d_ws, size_t ws_size,
                              hipStream_t stream) {
  (void)in_sizes; (void)n_in; (void)out_size; (void)ws_size;
  const int*   eidx     = (const int*)  d_in[0];
  const float* pos      = (const float*)d_in[1];
  // d_in[2] = atomic_numbers (only its length is used by the reference)
  const float* feats0   = (const float*)d_in[3];
  const float* feats1   = (const float*)d_in[4];
  const float* feats2   = (const float*)d_in[5];
  const float* bessel_w = (const float*)d_in[6];
  const float* rbf_w    = (const float*)d_in[7];
  const float* rbf_b    = (const float*)d_in[8];
  const float* lin1_w   = (const float*)d_in[9];
  const float* lin1_b   = (const float*)d_in[10];
  const float* lin2_w   = (const float*)d_in[11];
  const float* lin2_b   = (const float*)d_in[12];
  const float* act_w    = (const float*)d_in[13];
  const float* act_b    = (const float*)d_in[14];

  float* node_msg = (float*)d_ws;               // 20000*16*13 f32 = 16.64 MB
  float* out0 = (float*)d_out;
  float* out1 = out0 + N_ATOMS * NCH;
  float* out2 = out1 + N_ATOMS * NCH * 3;

  const int msg_elems = N_ATOMS * NCH * NSLOT;
  zero_ws_kernel<<<(msg_elems + 255) / 256, 256, 0, stream>>>(node_msg, msg_elems);

  // 16 edges per wave, 8 waves per block -> 128 edges/block, 2500 blocks exact
  edge_kernel<<<N_EDGES / 128, 256, 0, stream>>>(
      eidx, pos, feats0, feats1, feats2,
      bessel_w, rbf_w, rbf_b, lin1_w, lin1_b, node_msg);

  // 16 atoms per wave, 8 waves per block
  const int node_tiles = (N_ATOMS + 15) / 16;
  node_kernel<<<(node_tiles + 7) / 8, 256, 0, stream>>>(
      feats0, feats1, feats2, lin2_w, lin2_b, act_w, act_b,
      node_msg, out0, out1, out2);
}